// AttentionAggregationV2_23424751632786
// MI455X (gfx1250) — compile-verified
//
#include <hip/hip_runtime.h>
#include <hip/hip_bf16.h>
#include <stdint.h>

// Problem constants (match the reference)
#define NUM_HEADS 8
#define N_NODES   50000
#define N_EDGES   800000
#define VDIM      320          // fused value dim: 128 (8x16) + 192 (8x24)
#define TILE      32           // edges per LDS tile (40KB per buffer)
#define TPW       4            // tiles per workgroup (double-buffered TDM pipeline)

typedef __attribute__((ext_vector_type(4))) unsigned int uint32x4_t_;
typedef __attribute__((ext_vector_type(8))) int          int32x8_t_;
typedef __attribute__((ext_vector_type(4))) int          int32x4_t_;

// fused column -> head mapping (identical on input and output side)
__device__ __forceinline__ int head_of_col(int c) {
    return (c < 128) ? (c >> 4) : ((c - 128) / 24);
}

// Deterministic-result float atomic max via integer atomics (sign-split trick).
// Buffer must be initialized to -inf (0xFF800000).
__device__ __forceinline__ void atomic_max_float(float* addr, float v) {
    if (v >= 0.0f) {
        atomicMax((int*)addr, __float_as_int(v));
    } else {
        atomicMin((unsigned int*)addr, (unsigned int)__float_as_int(v));
    }
}

// Issue one TDM 2D-tile DMA: value rows [0,rows) x 320 f32 -> LDS (wave-uniform args)
__device__ __forceinline__ void tdm_load_tile(const float* gsrc, const float* lds_dst,
                                              int rows) {
    unsigned long long gaddr = (unsigned long long)(uintptr_t)gsrc;
    unsigned lds_base = (unsigned)(uintptr_t)lds_dst;  // low 32 bits = LDS offset

    uint32x4_t_ g0;
    g0.x = 1u;                                          // count=1 valid descriptor
    g0.y = lds_base;                                    // lds_addr
    g0.z = (unsigned)(gaddr & 0xFFFFFFFFull);           // global_addr[31:0]
    g0.w = (unsigned)((gaddr >> 32) & 0x1FFFFFFull)     // global_addr[56:32]
         | (2u << 30);                                  // type = 2 ("image")

    int32x8_t_ g1;
    g1[0] = (2 << 16);                                  // workgroup_mask=0, data_size=4B
    g1[1] = (int)(((unsigned)VDIM & 0xFFFFu) << 16);    // tensor_dim0 lo16 = 320
    g1[2] = (int)(((unsigned)rows & 0xFFFFu) << 16);    // dim0 hi=0 | tensor_dim1 lo16
    g1[3] = (int)((unsigned)VDIM << 16);                // dim1 hi=0 | tile_dim0 = 320
    g1[4] = TILE;                                       // tile_dim1=TILE, tile_dim2=0
    g1[5] = VDIM;                                       // tensor_dim0_stride lo32 = 320
    g1[6] = 0;
    g1[7] = 0;

    int32x4_t_ gz4 = {0, 0, 0, 0};                      // groups 2/3 unused (2D tensor)
    int32x8_t_ gz8 = {0, 0, 0, 0, 0, 0, 0, 0};
    __builtin_amdgcn_tensor_load_to_lds(g0, g1, gz4, gz4, gz8, 0);
}

// ---------------------------------------------------------------------------
// K2 (placed first so the disasm snippet shows it): per-tile aggregation with
// a double-buffered TDM pipeline.
//   iter i:  wave0 issues DMA for tile i+1 into buf[(i+1)&1]
//            threads 0..255 compute a = exp(w - m[dst]) for tile i + build s
//            wave0 waits tensorcnt<=1 (tile i landed; tile i+1 in flight)
//            320 threads (one per fused column) scatter a*v from buf[i&1]
// ---------------------------------------------------------------------------
__global__ __launch_bounds__(VDIM) void k_agg(
        const float* __restrict__ value,
        const float* __restrict__ ew,
        const float* __restrict__ cutoff,
        const int*   __restrict__ dst,
        const float* __restrict__ mbuf,
        float* __restrict__ sbuf,
        float* __restrict__ out) {

    __shared__ float v_tile[2][TILE * VDIM];     // 2 x 40 KB value buffers
    __shared__ float a_tile[TILE][NUM_HEADS];    // softmax numerators (tile i only)
    __shared__ int   dst_tile[TILE];

    const int base = blockIdx.x * (TILE * TPW);
    if (base >= N_EDGES) return;
    const int ntiles = min(TPW, (N_EDGES - base + TILE - 1) / TILE);
    const int t = threadIdx.x;
    const int c = t;                             // 0..319, blockDim.x == VDIM
    const int hc = head_of_col(c);

    // prologue: DMA tile 0
    if (t < 32) {
        tdm_load_tile(value + (size_t)base * VDIM, &v_tile[0][0],
                      min(TILE, N_EDGES - base));
    }

    for (int i = 0; i < ntiles; ++i) {
        const int e0   = base + i * TILE;
        const int rows = min(TILE, N_EDGES - e0);
        const bool has_next = (i + 1 < ntiles);

        // issue next tile's DMA into the other buffer (consumed in the previous iteration)
        if ((t < 32) && has_next) {
            const int e1 = e0 + TILE;
            tdm_load_tile(value + (size_t)e1 * VDIM, &v_tile[(i + 1) & 1][0],
                          min(TILE, N_EDGES - e1));
        }

        // softmax numerators for tile i (independent of the value tile)
        if (t < TILE * NUM_HEADS) {
            int r = t >> 3;
            int h = t & 7;
            int e = e0 + r;
            if (r < rows) {
                int dn = dst[e];
                if (h == 0) dst_tile[r] = dn;
                float w = cutoff[e] * ew[e * NUM_HEADS + h];
                float a = __expf(w - mbuf[dn * NUM_HEADS + h]);
                a_tile[r][h] = a;
                unsafeAtomicAdd(&sbuf[dn * NUM_HEADS + h], a);
            }
        }

        // TDM ops from one wave complete in order: <=1 outstanding => tile i landed
        if (t < 32) {
            if (has_next) __builtin_amdgcn_s_wait_tensorcnt(1);
            else          __builtin_amdgcn_s_wait_tensorcnt(0);
        }
        __syncthreads();

        // scatter: one thread per fused output column
        const float* vb = &v_tile[i & 1][0];
        for (int r = 0; r < rows; ++r) {
            float av = a_tile[r][hc] * vb[r * VDIM + c];   // conflict-free LDS read
            unsafeAtomicAdd(&out[(size_t)dst_tile[r] * VDIM + c], av);
        }
        __syncthreads();   // protect a_tile/dst_tile and v buffer reuse
    }
}

// ---------------------------------------------------------------------------
// K0: zero output accumulator, init per-(node,head) max to -inf and sum to 0
// ---------------------------------------------------------------------------
__global__ void k_init(float* __restrict__ out,
                       float* __restrict__ mbuf,
                       float* __restrict__ sbuf) {
    int idx = blockIdx.x * blockDim.x + threadIdx.x;
    if (idx < N_NODES * VDIM) out[idx] = 0.0f;
    if (idx < N_NODES * NUM_HEADS) {
        mbuf[idx] = __int_as_float(0xFF800000);  // -inf
        sbuf[idx] = 0.0f;
    }
}

// ---------------------------------------------------------------------------
// K1: segmented max: m[dst,h] = max_e cutoff[e]*ew[e,h]
// ---------------------------------------------------------------------------
__global__ void k_max(const float* __restrict__ ew,
                      const float* __restrict__ cutoff,
                      const int*   __restrict__ dst,
                      float* __restrict__ mbuf) {
    int idx = blockIdx.x * blockDim.x + threadIdx.x;
    if (idx >= N_EDGES * NUM_HEADS) return;
    int e = idx >> 3;
    int h = idx & 7;
    float w = cutoff[e] * ew[idx];          // ew is [E,8] row-major -> coalesced
    atomic_max_float(&mbuf[dst[e] * NUM_HEADS + h], w);
}

// ---------------------------------------------------------------------------
// K3: normalize: out[n,c] = (s>0) ? out[n,c]/s[n,head(c)] : 0
// ---------------------------------------------------------------------------
__global__ void k_finalize(float* __restrict__ out,
                           const float* __restrict__ sbuf) {
    int idx = blockIdx.x * blockDim.x + threadIdx.x;
    if (idx >= N_NODES * VDIM) return;
    int n = idx / VDIM;
    int c = idx - n * VDIM;
    float s = sbuf[n * NUM_HEADS + head_of_col(c)];
    float f = out[idx];
    out[idx] = (s > 0.0f) ? (f / s) : 0.0f;   // empty nodes -> exact 0 (matches ref)
}

// ---------------------------------------------------------------------------
extern "C" void kernel_launch(void* const* d_in, const int* in_sizes, int n_in,
                              void* d_out, int out_size, void* d_ws, size_t ws_size,
                              hipStream_t stream) {
    const float* value  = (const float*)d_in[0];   // [E, 320]
    const float* ew     = (const float*)d_in[1];   // [E, 8]
    const float* cutoff = (const float*)d_in[2];   // [E]
    const int*   eidx   = (const int*)d_in[3];     // [2, E]
    const int*   dst    = eidx + N_EDGES;          // second row = destinations

    float* out  = (float*)d_out;                   // [N, 320]
    float* mbuf = (float*)d_ws;                    // [N, 8] per-head max
    float* sbuf = mbuf + (size_t)N_NODES * NUM_HEADS;  // [N, 8] per-head sum

    k_init<<<(N_NODES * VDIM + 255) / 256, 256, 0, stream>>>(out, mbuf, sbuf);
    k_max<<<(N_EDGES * NUM_HEADS + 255) / 256, 256, 0, stream>>>(ew, cutoff, dst, mbuf);
    const int edges_per_wg = TILE * TPW;
    k_agg<<<(N_EDGES + edges_per_wg - 1) / edges_per_wg, VDIM, 0, stream>>>(
        value, ew, cutoff, dst, mbuf, sbuf, out);
    k_finalize<<<(N_NODES * VDIM + 255) / 256, 256, 0, stream>>>(out, sbuf);
}